// LDEPooling_51342039056682
// MI455X (gfx1250) — compile-verified
//
#include <hip/hip_runtime.h>

// LDE pooling, algebraically restructured into two fp32 WMMA GEMMs + softmax.
// B=8, C=256, K=64, T=1000.  All matrix math via V_WMMA_F32_16X16X4_F32
// (full fp32 precision; problem is bandwidth/latency bound, ~25 MB traffic,
// ~1.1us at 23.3 TB/s, so low-precision WMMA buys nothing and hurts softmax).
//
// Round-2 changes: removed divergent per-load EXEC guards in lde_weights
// (clamp t instead; garbage columns in the last tile are never consumed),
// and widened lde_pool's contiguous fragment loads to b64.

typedef float v2f __attribute__((ext_vector_type(2)));
typedef float v8f __attribute__((ext_vector_type(8)));

#define B_ 8
#define C_ 256
#define K_ 64
#define T_ 1000
#define NT_TILES 63            // ceil(T/16)

// workspace layout (floats):
#define WS_WBUF 0                       // weights (B,K,T)      512000
#define WS_CSQ  (B_*K_*T_)              // codesq[K]            64
#define WS_S1   (WS_CSQ + K_)           // S1 (B,K,C)           131072
// total = 643,136 floats ~ 2.6 MB

// ---------------------------------------------------------------------------
// Kernel 0: codesq[k] = sum_c codes[c,k]^2
// ---------------------------------------------------------------------------
__global__ __launch_bounds__(64) void lde_init(const float* __restrict__ codes,
                                               float* __restrict__ wsf) {
    int k = threadIdx.x;
    float s = 0.0f;
    for (int c = 0; c < C_; ++c) {
        float v = codes[c * K_ + k];
        s += v * v;
    }
    wsf[WS_CSQ + k] = s;
}

// ---------------------------------------------------------------------------
// Kernel 1: per (b, t-tile of 16):  dot[k,t] = sum_c codes[c,k]*x[b,c,t]
// via V_WMMA_F32_16X16X4_F32 (M=k tile, N=t tile, reduce c in steps of 4),
// then softmax over K per t-column -> weights into ws.
// 4 waves per block, wave w owns k-tile [16w,16w+16).
// ---------------------------------------------------------------------------
__global__ __launch_bounds__(128) void lde_weights(const float* __restrict__ x,
                                                   const float* __restrict__ codes,
                                                   const float* __restrict__ scales,
                                                   float* __restrict__ wsf) {
    const int b    = blockIdx.x / NT_TILES;
    const int tt   = blockIdx.x % NT_TILES;
    const int t0   = tt * 16;
    const int wave = threadIdx.x >> 5;
    const int lane = threadIdx.x & 31;
    const int half = lane >> 4;          // 0: reduce rows {0,1}, 1: rows {2,3}
    const int lr   = lane & 15;
    const int k0   = wave * 16;

    __shared__ float dot_s[4 * 16 * 16]; // [ktile][m][n]

    const float* xb = x + (size_t)b * C_ * T_;
    // Clamp instead of predicate: columns with t0+lr >= T produce garbage dot
    // values that the softmax stage never reads (it guards tg < T). This keeps
    // the inner loop branch-free (no EXEC save/restore around loads).
    const int tcl = (t0 + lr < T_) ? (t0 + lr) : (T_ - 1);

    v8f acc = {};
#pragma unroll 4
    for (int c = 0; c < C_; c += 4) {
        const int ca = c + 2 * half;
        v2f a, bv;
        // A (16x4, M=k, KK=c): VGPR0 = KK {0|2}, VGPR1 = KK {1|3}
        a.x = codes[(ca + 0) * K_ + k0 + lr];
        a.y = codes[(ca + 1) * K_ + k0 + lr];
        // B (4x16, KK=c, N=t): row-striped across lanes
        bv.x = xb[(ca + 0) * T_ + tcl];
        bv.y = xb[(ca + 1) * T_ + tcl];
        acc = __builtin_amdgcn_wmma_f32_16x16x4_f32(false, a, false, bv,
                                                    (short)0, acc, false, false);
    }
    // C/D layout: acc[r] = dot[m = r + 8*half][n = lr]
    for (int r = 0; r < 8; ++r)
        dot_s[wave * 256 + (r + 8 * half) * 16 + lr] = acc[r];
    __syncthreads();

    // softmax over K=64 for each of the 16 t-columns (threads 0..15),
    // fixed-order summation -> deterministic.
    if (threadIdx.x < 16) {
        const int tc = threadIdx.x;
        const int tg = t0 + tc;
        if (tg < T_) {
            float xs = 0.0f;
            for (int c = 0; c < C_; ++c) {
                float v = xb[c * T_ + tg];
                xs += v * v;
            }
            float mx = -3.4e38f;
            for (int k = 0; k < K_; ++k) {
                float d  = dot_s[(k >> 4) * 256 + (k & 15) * 16 + tc];
                float r2 = xs - 2.0f * d + wsf[WS_CSQ + k];
                mx = fmaxf(mx, -scales[k] * r2);
            }
            float sum = 0.0f;
            for (int k = 0; k < K_; ++k) {
                float d  = dot_s[(k >> 4) * 256 + (k & 15) * 16 + tc];
                float r2 = xs - 2.0f * d + wsf[WS_CSQ + k];
                sum += __expf(-scales[k] * r2 - mx);
            }
            const float inv = 1.0f / sum;
            for (int k = 0; k < K_; ++k) {
                float d  = dot_s[(k >> 4) * 256 + (k & 15) * 16 + tc];
                float r2 = xs - 2.0f * d + wsf[WS_CSQ + k];
                wsf[WS_WBUF + ((size_t)b * K_ + k) * T_ + tg] =
                    __expf(-scales[k] * r2 - mx) * inv;
            }
        }
    }
}

// ---------------------------------------------------------------------------
// Kernel 2: S1[b,k,c] = sum_t w[b,k,t] * x[b,c,t]
// One 16x16 (k x c) tile per wave, reduce T=1000 in steps of 4 (250 WMMAs).
// Fragment pairs are contiguous & 8B-aligned -> b64 loads.
// 512 tiles total = 128 blocks x 4 waves.
// ---------------------------------------------------------------------------
__global__ __launch_bounds__(128) void lde_pool(const float* __restrict__ x,
                                                float* __restrict__ wsf) {
    const int wave = threadIdx.x >> 5;
    const int lane = threadIdx.x & 31;
    const int half = lane >> 4;
    const int lr   = lane & 15;
    const int tile = blockIdx.x * 4 + wave;   // 0..511
    const int b    = tile >> 6;
    const int kt   = (tile >> 4) & 3;
    const int ct   = tile & 15;
    const int k0   = kt * 16, c0 = ct * 16;

    const float* xb = x + (size_t)b * C_ * T_;
    const float* wb = wsf + WS_WBUF + (size_t)b * K_ * T_;

    // row base offsets (element index is even at every access: rows are
    // multiples of T=1000 and ta is even -> 8-byte alignment for v2f loads)
    const float* arow = wb + (k0 + lr) * T_;
    const float* brow = xb + (c0 + lr) * T_;

    v8f acc = {};
#pragma unroll 4
    for (int t = 0; t < T_; t += 4) {       // T divisible by 4 exactly
        const int ta = t + 2 * half;
        v2f a  = *(const v2f*)(arow + ta);  // A: M=k, KK={ta, ta+1}
        v2f bv = *(const v2f*)(brow + ta);  // B: KK={ta, ta+1}, N=c
        acc = __builtin_amdgcn_wmma_f32_16x16x4_f32(false, a, false, bv,
                                                    (short)0, acc, false, false);
    }
    float* S1 = wsf + WS_S1;
    for (int r = 0; r < 8; ++r) {
        const int k = k0 + r + 8 * half;
        const int c = c0 + lr;
        S1[((size_t)b * K_ + k) * C_ + c] = acc[r];
    }
}

// ---------------------------------------------------------------------------
// Kernel 3: per (b,k):  sw = sum_t w;  e_c = (S1 - codes[c,k]*sw)/T;
// L2-normalize over c; write (B,K,C) output. Fixed-order tree reductions.
// ---------------------------------------------------------------------------
__global__ __launch_bounds__(256) void lde_finalize(const float* __restrict__ codes,
                                                    const float* __restrict__ wsf,
                                                    float* __restrict__ out) {
    const int b = blockIdx.x >> 6;
    const int k = blockIdx.x & 63;
    const int c = threadIdx.x;
    __shared__ float red[256];
    __shared__ float sw_s, invn_s;

    // sw = sum_t w[b,k,t]
    const float* wrow = wsf + WS_WBUF + ((size_t)b * K_ + k) * T_;
    float s = 0.0f;
    for (int t = c; t < T_; t += 256) s += wrow[t];
    red[c] = s;
    __syncthreads();
    for (int st = 128; st > 0; st >>= 1) {
        if (c < st) red[c] += red[c + st];
        __syncthreads();
    }
    if (c == 0) sw_s = red[0];
    __syncthreads();

    const float e = (wsf[WS_S1 + ((size_t)b * K_ + k) * C_ + c]
                     - codes[c * K_ + k] * sw_s) * (1.0f / (float)T_);
    red[c] = e * e;
    __syncthreads();
    for (int st = 128; st > 0; st >>= 1) {
        if (c < st) red[c] += red[c + st];
        __syncthreads();
    }
    if (c == 0) {
        float n = __builtin_sqrtf(red[0]);
        invn_s = 1.0f / fmaxf(n, 1e-12f);
    }
    __syncthreads();
    out[((size_t)b * K_ + k) * C_ + c] = e * invn_s;
}

// ---------------------------------------------------------------------------
extern "C" void kernel_launch(void* const* d_in, const int* in_sizes, int n_in,
                              void* d_out, int out_size, void* d_ws, size_t ws_size,
                              hipStream_t stream) {
    const float* x      = (const float*)d_in[0];   // (B,C,T)
    const float* codes  = (const float*)d_in[1];   // (C,K)
    const float* scales = (const float*)d_in[2];   // (K,)
    float* out = (float*)d_out;                    // (B,K,C)
    float* wsf = (float*)d_ws;                     // ~2.6 MB used

    lde_init    <<<1,              64,  0, stream>>>(codes, wsf);
    lde_weights <<<B_ * NT_TILES,  128, 0, stream>>>(x, codes, scales, wsf);
    lde_pool    <<<128,            128, 0, stream>>>(x, wsf);
    lde_finalize<<<B_ * K_,        256, 0, stream>>>(codes, wsf, out);
}